// PINSAN_Network_53523882443208
// MI455X (gfx1250) — compile-verified
//
#include <hip/hip_runtime.h>

#define B_TOTAL 65536
#define N_AGENTS 4096
#define N_SLOTS 16
#define DIM 128
#define LDSTR 132   // padded LDS row stride (floats), kills bank conflicts, keeps 16B align
#define EPSF 1e-12f

typedef __attribute__((ext_vector_type(2))) float v2f;
typedef __attribute__((ext_vector_type(8))) float v8f;

__device__ __forceinline__ float wred32(float v) {
#pragma unroll
  for (int m = 16; m >= 1; m >>= 1) v += __shfl_xor(v, m, 32);
  return v;
}

__device__ __forceinline__ float sigmoidf_(float x) { return 1.0f / (1.0f + expf(-x)); }

// Stage a 16x128 f32 tile from global into padded LDS using CDNA5 async
// LDS-direct loads (ASYNCcnt-tracked, no VGPR round trip).
// 256 threads x 2 x 16B = 8KB. 16B chunks never cross a padded row boundary.
__device__ __forceinline__ void stage_tile_async(const float* __restrict__ src,
                                                 float* dstLds, int t) {
#pragma unroll
  for (int i = 0; i < 2; ++i) {
    const int flat = i * 1024 + t * 4;
    const float* gp = src + flat;
    float* lp = dstLds + (flat >> 7) * LDSTR + (flat & 127);
    const unsigned lds_off = (unsigned)(uintptr_t)lp;  // flat addr[31:0] == LDS offset
    asm volatile("global_load_async_to_lds_b128 %0, %1, off"
                 :
                 : "v"(lds_off), "v"((unsigned long long)(uintptr_t)gp)
                 : "memory");
  }
  asm volatile("s_wait_asynccnt 0x0" ::: "memory");
}

// Out[16][LDSTR] = act( Xlds[16][LDSTR] @ W^T + bias ), W row-major [128][128].
// One wave per 16-column tile (block = 256 threads = 8 waves covers all 128 cols).
// ACT: 0=none, 1=sigmoid, 2=tanh, 3=exact gelu
template <int ACT>
__device__ __forceinline__ void gemm_tile(const float* Xlds, const float* __restrict__ W,
                                          const float* __restrict__ bias, float* Out) {
  const int lane = threadIdx.x & 31;
  const int wave = threadIdx.x >> 5;
  const int n0 = wave * 16;
  const int m = lane & 15;            // A: row M ; B: col N ; C: col N
  const int kh = (lane >> 4) << 1;    // K split: lanes 0-15 -> {0,1}, 16-31 -> {2,3}
  const float* xr = Xlds + m * LDSTR + kh;
  const float* wr = W + (size_t)(n0 + m) * DIM + kh;  // B[k][n] = W[n][k]
  v8f c = {0.f, 0.f, 0.f, 0.f, 0.f, 0.f, 0.f, 0.f};
#pragma unroll
  for (int k0 = 0; k0 < DIM; k0 += 4) {
    v2f a, b;
    a.x = xr[k0];
    a.y = xr[k0 + 1];
    b.x = wr[k0];
    b.y = wr[k0 + 1];
    c = __builtin_amdgcn_wmma_f32_16x16x4_f32(false, a, false, b, (short)0, c, false, false);
  }
  const float bn = bias[n0 + m];
  const int rbase = (lane >> 4) << 3;  // C: VGPR r -> M = r (+8 for high lanes)
#pragma unroll
  for (int r = 0; r < 8; ++r) {
    float v = c[r] + bn;
    if (ACT == 1) v = sigmoidf_(v);
    if (ACT == 2) v = tanhf(v);
    if (ACT == 3) v = 0.5f * v * (1.0f + erff(v * 0.70710678118654752f));
    Out[(rbase + r) * LDSTR + n0 + m] = v;
  }
}

// Cosine-attention softmax weights over 16 slots; pqrow = LDS row (128 f32),
// mem = agent's [16][128] slab. Whole wave cooperates; all lanes get w[0..15].
__device__ __forceinline__ void attn_weights(const float* pqrow, const float* __restrict__ mem,
                                             int lane, float* w) {
  const float scale = 0.08838834764831845f;  // 1/sqrt(128)
  const float4 pq = *(const float4*)(pqrow + lane * 4);
  const float qss = wred32(pq.x * pq.x + pq.y * pq.y + pq.z * pq.z + pq.w * pq.w);
  const float qinv = 1.0f / fmaxf(sqrtf(qss), EPSF);
  float sc[N_SLOTS];
#pragma unroll
  for (int s = 0; s < N_SLOTS; ++s) {
    const float4 m4 = *(const float4*)(mem + s * DIM + lane * 4);
    float d = pq.x * m4.x + pq.y * m4.y + pq.z * m4.z + pq.w * m4.w;
    float ss = m4.x * m4.x + m4.y * m4.y + m4.z * m4.z + m4.w * m4.w;
    d = wred32(d);
    ss = wred32(ss);
    sc[s] = d * qinv * (1.0f / fmaxf(sqrtf(ss), EPSF)) * scale;
  }
  float mx = sc[0];
#pragma unroll
  for (int s = 1; s < N_SLOTS; ++s) mx = fmaxf(mx, sc[s]);
  float sum = 0.f;
#pragma unroll
  for (int s = 0; s < N_SLOTS; ++s) { w[s] = expf(sc[s] - mx); sum += w[s]; }
  const float inv = 1.0f / sum;
#pragma unroll
  for (int s = 0; s < N_SLOTS; ++s) w[s] *= inv;
}

// ---------------- READ PATH: proj -> attention -> fused proj -> gelu -> LN ----
__global__ __launch_bounds__(256) void pinsan_read_kernel(
    const float* __restrict__ queries, const float* __restrict__ memory,
    const float* __restrict__ Wq, const float* __restrict__ bq,
    const float* __restrict__ Wf, const float* __restrict__ bf,
    const float* __restrict__ ln_g, const float* __restrict__ ln_b,
    const int* __restrict__ aidx, float* __restrict__ out) {
  __shared__ float X[16 * LDSTR];
  __shared__ float PQ[16 * LDSTR];
  __shared__ float RD[16 * LDSTR];
  const int t = threadIdx.x;
  const int b0 = blockIdx.x * 16;
  stage_tile_async(queries + (size_t)b0 * DIM, X, t);
  __syncthreads();
  gemm_tile<0>(X, Wq, bq, PQ);  // pq = q @ Wq^T + bq
  __syncthreads();
  const int lane = t & 31, wave = t >> 5;
#pragma unroll
  for (int rr = 0; rr < 2; ++rr) {
    const int row = wave * 2 + rr;
    const int b = b0 + row;
    const float* mem = memory + (size_t)aidx[b] * N_SLOTS * DIM;
    float w[N_SLOTS];
    attn_weights(PQ + row * LDSTR, mem, lane, w);
    float4 acc = {0.f, 0.f, 0.f, 0.f};
#pragma unroll
    for (int s = 0; s < N_SLOTS; ++s) {
      const float4 m4 = *(const float4*)(mem + s * DIM + lane * 4);
      acc.x += w[s] * m4.x; acc.y += w[s] * m4.y;
      acc.z += w[s] * m4.z; acc.w += w[s] * m4.w;
    }
    *(float4*)(RD + row * LDSTR + lane * 4) = acc;
  }
  __syncthreads();
  gemm_tile<3>(RD, Wf, bf, X);  // fused = gelu(read @ Wf^T + bf) -> reuse X
  __syncthreads();
#pragma unroll
  for (int rr = 0; rr < 2; ++rr) {  // layernorm per row
    const int row = wave * 2 + rr;
    const float4 f = *(const float4*)(X + row * LDSTR + lane * 4);
    const float mu = wred32(f.x + f.y + f.z + f.w) * (1.0f / 128.0f);
    const float d0 = f.x - mu, d1 = f.y - mu, d2 = f.z - mu, d3 = f.w - mu;
    const float var = wred32(d0 * d0 + d1 * d1 + d2 * d2 + d3 * d3) * (1.0f / 128.0f);
    const float rstd = rsqrtf(var + 1e-5f);
    const float4 g4 = *(const float4*)(ln_g + lane * 4);
    const float4 b4 = *(const float4*)(ln_b + lane * 4);
    float4 o;
    o.x = d0 * rstd * g4.x + b4.x;
    o.y = d1 * rstd * g4.y + b4.y;
    o.z = d2 * rstd * g4.z + b4.z;
    o.w = d3 * rstd * g4.w + b4.w;
    *(float4*)(out + (size_t)(b0 + row) * DIM + lane * 4) = o;
  }
}

// ---------------- WRITE PATH part 1: projections + attention weights ---------
__global__ __launch_bounds__(256) void pinsan_write_kernel(
    const float* __restrict__ wq_in, const float* __restrict__ memory,
    const float* __restrict__ Wq, const float* __restrict__ bq,
    const float* __restrict__ We, const float* __restrict__ be,
    const float* __restrict__ Wa, const float* __restrict__ ba,
    const float* __restrict__ Wg, const float* __restrict__ bg,
    const int* __restrict__ aidx,
    float* __restrict__ w_ws, float* __restrict__ ev_ws, float* __restrict__ av_ws) {
  __shared__ float X[16 * LDSTR];
  __shared__ float PQ[16 * LDSTR];
  __shared__ float EW[16 * LDSTR];
  __shared__ float AW[16 * LDSTR];
  const int t = threadIdx.x;
  const int b0 = blockIdx.x * 16;
  stage_tile_async(wq_in + (size_t)b0 * DIM, X, t);
  __syncthreads();
  gemm_tile<0>(X, Wq, bq, PQ);  // write pq
  gemm_tile<1>(X, We, be, EW);  // erase = sigmoid(...)
  gemm_tile<2>(X, Wa, ba, AW);  // add = tanh(...)
  __syncthreads();
  const int lane = t & 31, wave = t >> 5;
#pragma unroll
  for (int rr = 0; rr < 2; ++rr) {
    const int row = wave * 2 + rr;
    const int b = b0 + row;
    // intensity = sigmoid(wq . Wg + bg)
    const float4 xq = *(const float4*)(X + row * LDSTR + lane * 4);
    const float4 wg = *(const float4*)(Wg + lane * 4);
    const float gd = wred32(xq.x * wg.x + xq.y * wg.y + xq.z * wg.z + xq.w * wg.w) + bg[0];
    const float gi = sigmoidf_(gd);
    float w[N_SLOTS];
    attn_weights(PQ + row * LDSTR, memory + (size_t)aidx[b] * N_SLOTS * DIM, lane, w);
    if (lane < N_SLOTS) w_ws[(size_t)b * N_SLOTS + lane] = w[lane];
    float4 e4 = *(const float4*)(EW + row * LDSTR + lane * 4);
    float4 a4 = *(const float4*)(AW + row * LDSTR + lane * 4);
    e4.x *= gi; e4.y *= gi; e4.z *= gi; e4.w *= gi;
    a4.x *= gi; a4.y *= gi; a4.z *= gi; a4.w *= gi;
    *(float4*)(ev_ws + (size_t)b * DIM + lane * 4) = e4;
    *(float4*)(av_ws + (size_t)b * DIM + lane * 4) = a4;
  }
}

// ---------------- WRITE PATH part 2: deterministic per-agent segment sum -----
#define APB 2  // agents per block: 2*16*128*2 floats = 32KB LDS accumulators
__global__ __launch_bounds__(256) void pinsan_agent_kernel(
    const float* __restrict__ memory, const int* __restrict__ aidx,
    const float* __restrict__ w_ws, const float* __restrict__ ev_ws,
    const float* __restrict__ av_ws, float* __restrict__ out_mem) {
  __shared__ float E[APB * N_SLOTS * DIM];
  __shared__ float A[APB * N_SLOTS * DIM];
  __shared__ int ilds[256];
  __shared__ unsigned masks[8];
  __shared__ int cnt[APB];
  __shared__ float rsum[N_SLOTS];
  const int t = threadIdx.x;
  const int a0 = blockIdx.x * APB;
  for (int i = t; i < APB * N_SLOTS * DIM; i += 256) { E[i] = 0.f; A[i] = 0.f; }
  if (t < APB) cnt[t] = 0;
  __syncthreads();
  const int d = t & 127;
  const int sh = t >> 7;  // 0 or 1: this thread handles slots sh, sh+2, ..., sh+14
  for (int chunk = 0; chunk < B_TOTAL; chunk += 256) {
    const int my = aidx[chunk + t];
    ilds[t] = my;
    const bool match = (my >= a0) && (my < a0 + APB);
    const unsigned bm = (unsigned)__ballot(match);
    if ((t & 31) == 0) masks[t >> 5] = bm;
    __syncthreads();
    for (int wv = 0; wv < 8; ++wv) {
      unsigned mm = masks[wv];
      while (mm) {  // uniform across block; items processed in ascending order
        const int bit = __ffs(mm) - 1;
        mm &= mm - 1;
        const int i = wv * 32 + bit;
        const int item = chunk + i;
        const int al = ilds[i] - a0;
        if (t == 0) cnt[al]++;
        const float ev = ev_ws[(size_t)item * DIM + d];
        const float av = av_ws[(size_t)item * DIM + d];
#pragma unroll
        for (int k = 0; k < 8; ++k) {
          const int s = sh + k * 2;
          const float ws = w_ws[(size_t)item * N_SLOTS + s];
          E[(al * N_SLOTS + s) * DIM + d] += ws * ev;
          A[(al * N_SLOTS + s) * DIM + d] += ws * av;
        }
      }
    }
    __syncthreads();
  }
  // finalize each owned agent
  for (int al = 0; al < APB; ++al) {
    const int a = a0 + al;
    const float* mrow = memory + (size_t)a * N_SLOTS * DIM;
    float* orow = out_mem + (size_t)a * N_SLOTS * DIM;
    const bool present = (cnt[al] > 0);
    if (present) {
#pragma unroll
      for (int k = 0; k < 8; ++k) {  // nv = mem*(1-E)+A, stored back into E
        const int s = sh + k * 2;
        const int i = (al * N_SLOTS + s) * DIM + d;
        E[i] = mrow[s * DIM + d] * (1.0f - E[i]) + A[i];
      }
    }
    __syncthreads();
    if (present) {  // deterministic per-slot sum of squares (16 groups of 16 lanes)
      const int s = t >> 4, j = t & 15;
      float acc = 0.f;
#pragma unroll
      for (int q = 0; q < 8; ++q) {
        const float v = E[(al * N_SLOTS + s) * DIM + j * 8 + q];
        acc += v * v;
      }
#pragma unroll
      for (int msk = 8; msk >= 1; msk >>= 1) acc += __shfl_xor(acc, msk, 32);
      if (j == 0) rsum[s] = acc;
    }
    __syncthreads();
    if (present) {
#pragma unroll
      for (int k = 0; k < 8; ++k) {
        const int s = sh + k * 2;
        const float inv = 1.0f / fmaxf(sqrtf(rsum[s]), EPSF);
        orow[s * DIM + d] = E[(al * N_SLOTS + s) * DIM + d] * inv;
      }
    } else {
      for (int i = t; i < N_SLOTS * DIM; i += 256) orow[i] = mrow[i];
    }
    __syncthreads();
  }
}

extern "C" void kernel_launch(void* const* d_in, const int* in_sizes, int n_in,
                              void* d_out, int out_size, void* d_ws, size_t ws_size,
                              hipStream_t stream) {
  (void)in_sizes; (void)n_in; (void)out_size; (void)ws_size;
  const float* queries  = (const float*)d_in[0];
  const float* wqueries = (const float*)d_in[1];
  const float* memory   = (const float*)d_in[2];
  const float* Wq = (const float*)d_in[3];
  const float* bq = (const float*)d_in[4];
  const float* We = (const float*)d_in[5];
  const float* be = (const float*)d_in[6];
  const float* Wa = (const float*)d_in[7];
  const float* ba = (const float*)d_in[8];
  const float* Wf = (const float*)d_in[9];
  const float* bf = (const float*)d_in[10];
  const float* ln_g = (const float*)d_in[11];
  const float* ln_b = (const float*)d_in[12];
  const float* Wg = (const float*)d_in[13];
  const float* bg = (const float*)d_in[14];
  const int* aidx = (const int*)d_in[15];

  float* out_read = (float*)d_out;                         // [B, 128]
  float* out_mem  = out_read + (size_t)B_TOTAL * DIM;      // [4096, 16, 128]

  float* w_ws  = (float*)d_ws;                             // [B, 16]
  float* ev_ws = w_ws + (size_t)B_TOTAL * N_SLOTS;         // [B, 128]
  float* av_ws = ev_ws + (size_t)B_TOTAL * DIM;            // [B, 128]   total 68MB

  pinsan_read_kernel<<<B_TOTAL / 16, 256, 0, stream>>>(
      queries, memory, Wq, bq, Wf, bf, ln_g, ln_b, aidx, out_read);
  pinsan_write_kernel<<<B_TOTAL / 16, 256, 0, stream>>>(
      wqueries, memory, Wq, bq, We, be, Wa, ba, Wg, bg, aidx, w_ws, ev_ws, av_ws);
  pinsan_agent_kernel<<<N_AGENTS / APB, 256, 0, stream>>>(
      memory, aidx, w_ws, ev_ws, av_ws, out_mem);
}